// EdgewiseEnergySum_59777354826469
// MI455X (gfx1250) — compile-verified
//
#include <hip/hip_runtime.h>

// Edge-wise energy scale + segment-sum for MI455X (gfx1250, wave32).
// Memory-bound scatter-add: ~77 MB streamed reads -> ~3.3 us roofline @ 23.3 TB/s.

typedef float v4f __attribute__((ext_vector_type(4)));
typedef int   v4i __attribute__((ext_vector_type(4)));

#define BLOCK 256            // 8 wave32 waves per block
#define SCATTER_BLOCKS 2048  // grid-stride; keeps all WGPs saturated
#define INV_SQRT_AVG_NEIGH 0.125f  // 1/sqrt(64)

__global__ __launch_bounds__(BLOCK) void zero_out_kernel(float* __restrict__ out, int n) {
    int i = (blockIdx.x * BLOCK + threadIdx.x) * 4;
    if (i + 3 < n) {
        *(v4f*)(out + i) = (v4f)0.0f;
    } else {
        for (; i < n; ++i) out[i] = 0.0f;   // at most 3 scalar writes, one thread
    }
}

__global__ __launch_bounds__(BLOCK) void edge_scatter_kernel(
    const float* __restrict__ eng,      // [n_edges]
    const int*   __restrict__ idx_c,    // [n_edges] edge center
    const int*   __restrict__ idx_n,    // [n_edges] edge neighbor
    const int*   __restrict__ species,  // [n_nodes], values 0..3
    const float* __restrict__ scales,   // [4*4]
    float*       __restrict__ out,      // [n_nodes]
    int n_edges)
{
    // Stage the 4x4 scale table (with 1/sqrt(64) folded in) into LDS once.
    __shared__ float s_scale[16];
    if (threadIdx.x < 16)
        s_scale[threadIdx.x] = scales[threadIdx.x] * INV_SQRT_AVG_NEIGH;
    __syncthreads();

    const int nvec   = n_edges >> 2;                 // float4 / int4 chunks
    const int stride = gridDim.x * BLOCK;

    const v4f* __restrict__ eng4 = (const v4f*)eng;
    const v4i* __restrict__ c4   = (const v4i*)idx_c;
    const v4i* __restrict__ n4   = (const v4i*)idx_n;

    for (int t = blockIdx.x * BLOCK + threadIdx.x; t < nvec; t += stride) {
        // Speculative prefetch one grid-stride ahead (global_prefetch_b8);
        // OOB translation failures are silently dropped.
        __builtin_prefetch(eng4 + t + stride, 0, 0);
        __builtin_prefetch(c4   + t + stride, 0, 0);
        __builtin_prefetch(n4   + t + stride, 0, 0);

        // 128-bit non-temporal streaming loads (th:TH_LOAD_NT): this 77 MB
        // stream is read-once; don't evict the hot species/output lines.
        v4f e = __builtin_nontemporal_load(eng4 + t);
        v4i c = __builtin_nontemporal_load(c4 + t);
        v4i n = __builtin_nontemporal_load(n4 + t);

#pragma unroll
        for (int k = 0; k < 4; ++k) {
            const int ci = c[k];
            const int ni = n[k];
            // species array (400 KB) lives in L2/WGP$ — random gathers are cache hits
            const float s = s_scale[(species[ci] << 2) | species[ni]];
            // no-return global_atomic_add_f32: L2 atomic units do the RMW
            atomicAdd(out + ci, e[k] * s);
        }
    }

    // Scalar tail for n_edges % 4 != 0 (zero for the 6.4M reference shape).
    const int rem = n_edges & 3;
    if (blockIdx.x == 0 && (int)threadIdx.x < rem) {
        const int ei = n_edges - rem + threadIdx.x;
        const int ci = idx_c[ei];
        const int ni = idx_n[ei];
        const float s = s_scale[(species[ci] << 2) | species[ni]];
        atomicAdd(out + ci, eng[ei] * s);
    }
}

extern "C" void kernel_launch(void* const* d_in, const int* in_sizes, int n_in,
                              void* d_out, int out_size, void* d_ws, size_t ws_size,
                              hipStream_t stream) {
    const float* eng     = (const float*)d_in[0];   // edge_eng  [E,1] f32
    const int*   idx     = (const int*)  d_in[1];   // edge_index [2,E] int
    const int*   species = (const int*)  d_in[2];   // species   [N,1] int
    const float* scales  = (const float*)d_in[3];   // per_edge_scales [4,4] f32
    float*       out     = (float*)d_out;           // atom_eng  [N,1] f32

    const int n_edges = in_sizes[0];
    const int n_nodes = out_size;

    const int*   idx_c = idx;
    const int*   idx_n = idx + n_edges;

    // 1) zero the (poisoned) output
    int zthreads = (n_nodes + 3) / 4;
    int zblocks  = (zthreads + BLOCK - 1) / BLOCK;
    zero_out_kernel<<<zblocks, BLOCK, 0, stream>>>(out, n_nodes);

    // 2) scaled scatter-add
    int nvec   = n_edges >> 2;
    int blocks = (nvec + BLOCK - 1) / BLOCK;
    if (blocks > SCATTER_BLOCKS) blocks = SCATTER_BLOCKS;
    if (blocks < 1) blocks = 1;
    edge_scatter_kernel<<<blocks, BLOCK, 0, stream>>>(
        eng, idx_c, idx_n, species, scales, out, n_edges);
}